// CapsuleLayer_25271587569923
// MI455X (gfx1250) — compile-verified
//
#include <hip/hip_runtime.h>

// CapsuleLayer dynamic routing, fully fused for gfx1250 (MI455X).
// One workgroup per (capsule c, batch-block of 16). Priors never touch HBM:
// recomputed on the fly via V_WMMA_F32_16X16X4_F32 with A = W_n^T (o x i),
// B = x-slice (i x 16 batches), accumulating s[o][b'] over n in the WMMA C
// operand. Routing state (logits) lives in LDS (16 x 1152 f32 = 72KB).

#define N_CAPS   10
#define BATCH    256
#define NROUTE   1152
#define IN_D     8
#define OUT_D    16
#define BB       16          // batch block (WMMA N dimension)
#define WAVES    8
#define NCHUNK   (NROUTE / WAVES)   // 144 route nodes per wave
#define ITERS    3

typedef __attribute__((ext_vector_type(2))) float v2f;
typedef __attribute__((ext_vector_type(8))) float v8f;

__global__ __launch_bounds__(256)
void caps_route_wmma(const float* __restrict__ x,      // [256][1152][8]
                     const float* __restrict__ w,      // [10][1152][8][16]
                     float* __restrict__ out)          // [10][256][16]
{
    extern __shared__ float lds[];
    float* logits = lds;                         // [BB][NROUTE]
    float* sacc   = lds + BB * NROUTE;           // [OUT_D][BB]  (o-major)
    float* scale  = sacc + OUT_D * BB;           // [BB]
    float* wred   = scale + BB;                  // [WAVES][BB]

    const int tid  = threadIdx.x;
    const int wave = tid >> 5;
    const int lane = tid & 31;
    const int half = lane >> 4;        // lane half selects K pair / M offset
    const int col  = lane & 15;        // A: row o ; B/C: column b'
    const int k0   = half * 2;         // K base for this lane's A/B elements

    const int bx = blockIdx.x;         // 160 blocks
    const int c  = bx >> 4;            // capsule 0..9
    const int b0 = (bx & 15) * BB;     // batch base

    const float* Wc = w + (size_t)c * NROUTE * IN_D * OUT_D;
    const float* Xb = x + (size_t)(b0 + col) * NROUTE * IN_D;  // this lane's batch row

    // zero logits
    for (int i = tid; i < BB * NROUTE; i += 256) logits[i] = 0.0f;
    __syncthreads();

    const int n0 = wave * NCHUNK;

    for (int it = 0; it < ITERS; ++it) {
        // ---------- softmax statistics over n, per batch col ----------
        float lmax = -3.4e38f;
        for (int j = 0; j < NCHUNK / 2; ++j) {
            int n = n0 + half + 2 * j;
            lmax = fmaxf(lmax, logits[col * NROUTE + n]);
        }
        lmax = fmaxf(lmax, __shfl_xor(lmax, 16, 32));
        if (lane < 16) wred[wave * 16 + col] = lmax;
        for (int i = tid; i < OUT_D * BB; i += 256) sacc[i] = 0.0f;
        __syncthreads();
        float gmax = -3.4e38f;
        for (int wv = 0; wv < WAVES; ++wv) gmax = fmaxf(gmax, wred[wv * 16 + col]);
        __syncthreads();
        float lsum = 0.0f;
        for (int j = 0; j < NCHUNK / 2; ++j) {
            int n = n0 + half + 2 * j;
            lsum += __expf(logits[col * NROUTE + n] - gmax);
        }
        lsum += __shfl_xor(lsum, 16, 32);
        if (lane < 16) wred[wave * 16 + col] = lsum;
        __syncthreads();
        float gsum = 0.0f;
        for (int wv = 0; wv < WAVES; ++wv) gsum += wred[wv * 16 + col];
        const float ginv = 1.0f / gsum;

        // ---------- s[o][b'] = sum_n p[b',n] * priors[n,b',o] via WMMA ----------
        v8f acc = {};
        for (int n = n0; n < n0 + NCHUNK; ++n) {
            float p = __expf(logits[col * NROUTE + n] - gmax) * ginv;
            const float* wp = Wc + ((size_t)n * IN_D + k0) * OUT_D + col;
            v2f a0; a0.x = wp[0];           a0.y = wp[OUT_D];        // i = k0, k0+1
            v2f a1; a1.x = wp[4 * OUT_D];   a1.y = wp[5 * OUT_D];    // i = k0+4, k0+5
            const float* xp = Xb + (size_t)n * IN_D + k0;
            v2f b0v; b0v.x = xp[0] * p;  b0v.y = xp[1] * p;
            v2f b1v; b1v.x = xp[4] * p;  b1v.y = xp[5] * p;
            acc = __builtin_amdgcn_wmma_f32_16x16x4_f32(false, a0, false, b0v,
                                                        (short)0, acc, false, false);
            acc = __builtin_amdgcn_wmma_f32_16x16x4_f32(false, a1, false, b1v,
                                                        (short)0, acc, false, false);
        }
        #pragma unroll
        for (int r = 0; r < 8; ++r) {
            int o = r + 8 * half;
            atomicAdd(&sacc[o * 16 + col], acc[r]);   // ds_add_f32 cross-wave reduce
        }
        __syncthreads();

        // ---------- squash: per-batch scale ----------
        if (tid < 16) {
            float sq = 0.0f;
            for (int o = 0; o < OUT_D; ++o) { float s = sacc[o * 16 + tid]; sq += s * s; }
            scale[tid] = (sq / (1.0f + sq)) * rsqrtf(sq);
        }
        __syncthreads();

        if (it == ITERS - 1) {
            // out[c][b0+b][o]
            int o = tid >> 4, b = tid & 15;
            out[((size_t)(c * BATCH + b0 + b)) * OUT_D + o] = sacc[o * 16 + b] * scale[b];
        } else {
            // ---------- delta: logits[b'][n] += sum_o priors[n,b',o] * v[o,b'] ----------
            float vreg[8];
            #pragma unroll
            for (int r = 0; r < 8; ++r)
                vreg[r] = sacc[(r + 8 * half) * 16 + col] * scale[col];
            for (int n = n0; n < n0 + NCHUNK; ++n) {
                const float* wp = Wc + ((size_t)n * IN_D + k0) * OUT_D + col;
                v2f a0; a0.x = wp[0];         a0.y = wp[OUT_D];
                v2f a1; a1.x = wp[4 * OUT_D]; a1.y = wp[5 * OUT_D];
                const float* xp = Xb + (size_t)n * IN_D + k0;
                v2f b0v; b0v.x = xp[0]; b0v.y = xp[1];
                v2f b1v; b1v.x = xp[4]; b1v.y = xp[5];
                v8f d = {};
                d = __builtin_amdgcn_wmma_f32_16x16x4_f32(false, a0, false, b0v,
                                                          (short)0, d, false, false);
                d = __builtin_amdgcn_wmma_f32_16x16x4_f32(false, a1, false, b1v,
                                                          (short)0, d, false, false);
                float partial = 0.0f;
                #pragma unroll
                for (int r = 0; r < 8; ++r) partial += d[r] * vreg[r];
                partial += __shfl_xor(partial, 16, 32);   // combine M=0..7 / M=8..15 halves
                if (lane < 16) logits[col * NROUTE + n] += partial;
            }
            __syncthreads();
        }
    }
}

extern "C" void kernel_launch(void* const* d_in, const int* in_sizes, int n_in,
                              void* d_out, int out_size, void* d_ws, size_t ws_size,
                              hipStream_t stream) {
    (void)in_sizes; (void)n_in; (void)d_ws; (void)ws_size; (void)out_size;
    const float* x = (const float*)d_in[0];
    const float* w = (const float*)d_in[1];
    float* out = (float*)d_out;

    const size_t lds_bytes =
        (size_t)(BB * NROUTE + OUT_D * BB + BB + WAVES * BB) * sizeof(float); // ~75.3 KB

    dim3 grid(N_CAPS * (BATCH / BB));   // 160 workgroups
    dim3 block(256);                    // 8 waves (wave32)
    caps_route_wmma<<<grid, block, lds_bytes, stream>>>(x, w, out);
}